// PC_shielded_electrostatics_17609365914540
// MI455X (gfx1250) — compile-verified
//
#include <hip/hip_runtime.h>
#include <hip/hip_bf16.h>

// ---------------------------------------------------------------------------
// PhysNet-style shielded electrostatics + segment_sum, MI455X (gfx1250).
//
// Memory-bound scatter/gather: 12 B/edge streamed + L2-resident gathers and
// fp32 atomics. WMMA is intentionally NOT used: there is no dense GEMM
// structure (random indices), and the 154 MB stream at 23.3 TB/s (~6.6 us)
// dominates the 0.32 GFLOP of edge math by orders of magnitude.
// CDNA5 paths used instead: wave32, b128 non-temporal streaming loads,
// global_prefetch_b8, hardware global_atomic_add_f32 (no CAS loop).
// ---------------------------------------------------------------------------

typedef float v4f __attribute__((ext_vector_type(4)));
typedef int   v4i __attribute__((ext_vector_type(4)));

#define KEHALF      7.199822675975274f
#define INV_SHORT   0.25f          // 1 / SHORT_CUTOFF(4.0)
#define INV_LC_SQ   0.01f          // 1 / (10*10)
#define TWO_OVER_LC 0.2f           // 2 / 10
#define LONG_CUTOFF 10.0f

__global__ __launch_bounds__(256) void pc_zero_out(float* __restrict__ out, int n) {
    int i = blockIdx.x * blockDim.x + threadIdx.x;
    if (i < n) out[i] = 0.0f;
}

__device__ __forceinline__ float edge_energy(float d, float qi, float qj) {
    // shielded distance sqrt(d^2 + 1)
    float dsh = __builtin_amdgcn_sqrtf(__builtin_fmaf(d, d, 1.0f));
    // poly6 switch: x = clamp(d/4, 0, 1); sw = 1 - x^3*(6x^2 - 15x + 10)
    float x  = __builtin_fmaxf(0.0f, __builtin_fminf(d * INV_SHORT, 1.0f));
    float p  = __builtin_fmaf(x, __builtin_fmaf(6.0f, x, -15.0f), 10.0f);
    float x3 = x * x * x;
    float sw  = 1.0f - x3 * p;
    float csw = 1.0f - sw;
    // damped/shifted Coulomb terms
    float E_ord = __builtin_amdgcn_rcpf(d)   + __builtin_fmaf(d,   INV_LC_SQ, -TWO_OVER_LC);
    float E_shd = __builtin_amdgcn_rcpf(dsh) + __builtin_fmaf(dsh, INV_LC_SQ, -TWO_OVER_LC);
    return KEHALF * qi * qj * __builtin_fmaf(csw, E_shd, sw * E_ord);
}

__global__ __launch_bounds__(256) void pc_shielded_electrostatics_kernel(
    const float* __restrict__ charges,
    const float* __restrict__ dist,
    const int*   __restrict__ idx_i,
    const int*   __restrict__ idx_j,
    float*       __restrict__ out,
    int n_edges)
{
    const int tid  = blockIdx.x * blockDim.x + threadIdx.x;
    const int base = tid * 4;

    if (base + 3 < n_edges) {
        // Prefetch the stream ahead (gfx1250 global_prefetch_b8, speculative NT).
        const int pf = base + 4 * 256 * 8;   // ~8 blocks ahead
        __builtin_prefetch(dist  + pf, 0, 0);
        __builtin_prefetch(idx_i + pf, 0, 0);
        __builtin_prefetch(idx_j + pf, 0, 0);

        // One b128 NT load per array: 4 edges per thread, read-once data.
        v4f dv = __builtin_nontemporal_load((const v4f*)(dist  + base));
        v4i iv = __builtin_nontemporal_load((const v4i*)(idx_i + base));
        v4i jv = __builtin_nontemporal_load((const v4i*)(idx_j + base));

#pragma unroll
        for (int k = 0; k < 4; ++k) {
            float d = dv[k];
            if (d <= LONG_CUTOFF) {            // beyond cutoff contributes exactly 0: skip atomic
                float qi = charges[iv[k]];     // L2-resident gathers (800 KB table)
                float qj = charges[jv[k]];
                float E  = edge_energy(d, qi, qj);
                unsafeAtomicAdd(&out[iv[k]], E);   // hw global_atomic_add_f32, no return
            }
        }
    } else {
        // scalar tail
        for (int e = base; e < n_edges; ++e) {
            float d = dist[e];
            if (d <= LONG_CUTOFF) {
                float E = edge_energy(d, charges[idx_i[e]], charges[idx_j[e]]);
                unsafeAtomicAdd(&out[idx_i[e]], E);
            }
        }
    }
}

extern "C" void kernel_launch(void* const* d_in, const int* in_sizes, int n_in,
                              void* d_out, int out_size, void* d_ws, size_t ws_size,
                              hipStream_t stream) {
    // setup_inputs() order: atomic_charges, distances, idx_i, idx_j, n_atoms
    const float* charges = (const float*)d_in[0];
    const float* dist    = (const float*)d_in[1];
    const int*   idx_i   = (const int*)  d_in[2];
    const int*   idx_j   = (const int*)  d_in[3];
    float*       out     = (float*)d_out;

    const int n_edges = in_sizes[1];
    const int n_atoms = out_size;

    // Zero the segment-sum accumulator (harness poisons d_out).
    {
        int blocks = (n_atoms + 255) / 256;
        pc_zero_out<<<blocks, 256, 0, stream>>>(out, n_atoms);
    }

    // 4 edges per thread, 256 threads (8 wave32) per block.
    {
        int threads = (n_edges + 3) / 4;
        int blocks  = (threads + 255) / 256;
        pc_shielded_electrostatics_kernel<<<blocks, 256, 0, stream>>>(
            charges, dist, idx_i, idx_j, out, n_edges);
    }
}